// Dense_RBS_density_31963146617515
// MI455X (gfx1250) — compile-verified
//
#include <hip/hip_runtime.h>
#include <math.h>

// Problem constants: dim = C(32,2) = 496 = 31*16, 15 RBS gates, batch 8.
#define DIM      496
#define MPAD     512         // M padded to 32 tiles so we can block 2x in M
#define NGATES   15
#define BATCH    8
#define NTILES   31          // 496 / 16  (N direction, unpadded)
#define MPAIRS   16          // 512 / 32  (M direction, 2-tile pairs)
#define TPB_GEMM 256         // 8 waves per block, 1 wave per 32x16 output block

typedef __attribute__((ext_vector_type(2))) float v2f;
typedef __attribute__((ext_vector_type(8))) float v8f;

__device__ __forceinline__ int pair_off(int i) {
    // offset of pairs starting with qubit i in combinations(range(32),2) order
    return i * 31 - (i * (i - 1)) / 2;
}
__device__ __forceinline__ int pair_idx(int i, int j) {  // requires i < j
    return pair_off(i) + (j - i - 1);
}

// ---------------------------------------------------------------------------
// Kernel 1: build U_tot = U_14 * ... * U_0 in workspace (MPAD x DIM f32,
// rows 496..511 are zero padding). Each gate g = (2g, 2g+1) is identity +
// 30 Givens rotations on row pairs (idx(c,2g), idx(c,2g+1)). Row pairs within
// a gate are disjoint; each thread owns one (pair, column) -> race free.
// Single workgroup; __syncthreads between the 15 sequential gates.
// ---------------------------------------------------------------------------
__global__ void rbs_build_u(const float* __restrict__ angles,
                            float* __restrict__ U) {
    const int total = MPAD * DIM;
    for (int t = threadIdx.x; t < total; t += blockDim.x) {
        int r = t / DIM, c = t % DIM;
        U[t] = (r == c) ? 1.0f : 0.0f;   // rows >= 496 have r != c -> zero
    }
    __syncthreads();

    for (int g = 0; g < NGATES; ++g) {
        const int a = 2 * g, b = 2 * g + 1;
        const float th = angles[g];
        const float ct = cosf(th);
        const float st = sinf(th);
        // U <- U_g * U : row_k' = c*row_k + s*row_kp ; row_kp' = -s*row_k + c*row_kp
        for (int t = threadIdx.x; t < 30 * DIM; t += blockDim.x) {
            const int oi  = t / DIM;                 // 0..29 -> "other" qubit
            const int col = t % DIM;
            const int c   = (oi < a) ? oi : oi + 2;  // skip qubits a, b=a+1
            const int k   = (c < a) ? pair_idx(c, a) : pair_idx(a, c);  // has a
            const int kp  = (c < b) ? pair_idx(c, b) : pair_idx(b, c);  // has b
            const float x = U[k  * DIM + col];
            const float y = U[kp * DIM + col];
            U[k  * DIM + col] =  ct * x + st * y;
            U[kp * DIM + col] = -st * x + ct * y;
        }
        __syncthreads();
    }
}

// ---------------------------------------------------------------------------
// Kernels 2/3: 32x16-block GEMM on V_WMMA_F32_16X16X4_F32 (full f32).
// Each wave owns TWO vertically adjacent 16x16 tiles (mt0, mt0+16) so the
// B-fragment is loaded once per k-step and reused by 2 WMMAs.
//   TRANSB=false : C[b] = A * B[b]        (T = U * rho), strided B loads
//   TRANSB=true  : C[b] = A[b] * B^T      (out = T * U^T), contiguous B loads
// GUARD=true clips stores to rows < 496 (M padding); guard is store-only so
// EXEC is all-ones at every WMMA.
// A-frag : lane holds A[m + (lane&15)][k + 2*(lane>>4) + {0,1}]  (float2)
// B-frag : VGPR0 rows K={0,2}, VGPR1 rows K={1,3}, N = lane&15
// D      : VGPR v -> row m + v + 8*(lane>>4), col nt + (lane&15)
// ---------------------------------------------------------------------------
template <bool TRANSB, bool GUARD>
__global__ void rbs_wmma_gemm(const float* __restrict__ Abase, int aStride,
                              const float* __restrict__ Bbase, int bStride,
                              float* __restrict__ Cbase, int cStride) {
    const int lane = threadIdx.x & 31;
    const int w    = blockIdx.x * (TPB_GEMM >> 5) + (threadIdx.x >> 5);

    const int batch = w / (MPAIRS * NTILES);
    const int rem   = w % (MPAIRS * NTILES);
    const int mt0   = (rem / NTILES) * 32;   // first of the two M tiles
    const int nt    = (rem % NTILES) * 16;

    const float* A = Abase + (size_t)batch * (size_t)aStride;
    const float* B = Bbase + (size_t)batch * (size_t)bStride;
    float*       C = Cbase + (size_t)batch * (size_t)cStride;

    const int half = lane >> 4;   // 0: lanes 0-15, 1: lanes 16-31
    const int l15  = lane & 15;

    const float* arow0 = A + (size_t)(mt0 + l15)      * DIM;
    const float* arow1 = A + (size_t)(mt0 + 16 + l15) * DIM;
    const float* brow  = TRANSB ? (B + (size_t)(nt + l15) * DIM) : nullptr;

    v8f acc0 = {};
    v8f acc1 = {};
#pragma unroll 2
    for (int k = 0; k < DIM; k += 4) {
        const int kk = k + 2 * half;
        v2f bf;
        if (TRANSB) {
            // B-frag element (K,N=n) = U[nt+n][k+K] -> contiguous float2
            bf.x = brow[kk];
            bf.y = brow[kk + 1];
        } else {
            bf.x = B[(size_t)kk       * DIM + nt + l15];
            bf.y = B[(size_t)(kk + 1) * DIM + nt + l15];
        }
        v2f a0, a1;
        a0.x = arow0[kk];  a0.y = arow0[kk + 1];
        a1.x = arow1[kk];  a1.y = arow1[kk + 1];
        acc0 = __builtin_amdgcn_wmma_f32_16x16x4_f32(
            false, a0, false, bf, (short)0, acc0, false, false);
        acc1 = __builtin_amdgcn_wmma_f32_16x16x4_f32(
            false, a1, false, bf, (short)0, acc1, false, false);
    }

#pragma unroll
    for (int v = 0; v < 8; ++v) {
        const int r0 = mt0 + v + 8 * half;
        const int r1 = r0 + 16;
        if (!GUARD || r0 < DIM)
            C[(size_t)r0 * DIM + nt + l15] = acc0[v];
        if (!GUARD || r1 < DIM)
            C[(size_t)r1 * DIM + nt + l15] = acc1[v];
    }
}

// ---------------------------------------------------------------------------
// Launch: build U (1 WG), then two L2-resident batched WMMA GEMMs.
// Workspace layout: [ U : 512*496 f32 | T : 8*512*496 f32 ]  (~9.1 MB)
// ---------------------------------------------------------------------------
extern "C" void kernel_launch(void* const* d_in, const int* in_sizes, int n_in,
                              void* d_out, int out_size, void* d_ws, size_t ws_size,
                              hipStream_t stream) {
    const float* rho    = (const float*)d_in[0];  // (8, 496, 496) f32
    const float* angles = (const float*)d_in[1];  // (15,) f32
    float* out = (float*)d_out;                   // (8, 496, 496) f32

    float* U = (float*)d_ws;                      // MPAD x DIM (pad rows zero)
    float* T = U + (size_t)MPAD * DIM;            // 8 x MPAD x DIM

    // 1) U_tot = product of 15 Givens-structured gate matrices
    rbs_build_u<<<1, 1024, 0, stream>>>(angles, U);

    // 2) T[b] = U * rho[b]   (M padded; pad rows of T become exact zeros)
    // 3) out[b] = T[b] * U^T (store-guard clips pad rows)
    const int nWaves  = BATCH * MPAIRS * NTILES;      // 3968 waves
    const int nBlocks = nWaves / (TPB_GEMM >> 5);     // 496 blocks exactly
    rbs_wmma_gemm<false, false><<<nBlocks, TPB_GEMM, 0, stream>>>(
        U, 0, rho, DIM * DIM, T, MPAD * DIM);
    rbs_wmma_gemm<true, true><<<nBlocks, TPB_GEMM, 0, stream>>>(
        T, MPAD * DIM, U, 0, out, DIM * DIM);
}